// RoIAlign_36034775613839
// MI455X (gfx1250) — compile-verified
//
#include <hip/hip_runtime.h>
#include <hip/hip_bf16.h>

// RoIAlign, MI455X (gfx1250, wave32).
// features: [4, 256, 200, 200] f32 (NCHW), rois: [R, 5] f32, out: [R, 256, 7, 7] f32.
//
// Strategy: gather-bound kernel. Per-roi sampling metadata (196 sample points,
// 4 clamped offsets + 4 validity-masked bilinear weights each) computed once per
// block into LDS and amortized over channels. The 196-sample -> 49-bin averaging
// is a real matmul out[16ch x 49bin] += V[16ch x 4pts] * B[4pts x 16bin] done
// with V_WMMA_F32_16X16X4_F32 (f32 end to end, B is a static 0/0.25 pattern
// synthesized in registers per the CDNA5 B-matrix VGPR layout).

typedef __attribute__((ext_vector_type(2))) float v2f;
typedef __attribute__((ext_vector_type(8))) float v8f;

#define FH 200
#define FW 200
#define FC 256
#define NPTS 196              // (7*2) * (7*2) sample points
#define NBIN 49               // 7*7 output bins

__global__ __launch_bounds__(256) void roialign_wmma_kernel(
    const float* __restrict__ feat,
    const float* __restrict__ rois,
    float* __restrict__ out)
{
    __shared__ int4   s_off[NPTS];   // 4 gather offsets (y*W+x), pre-clamped
    __shared__ float4 s_w[NPTS];     // 4 bilinear weights, pre-masked by validity

    const int r   = blockIdx.x;
    const int tid = threadIdx.x;

    // ---------------- Stage 1: per-roi sample descriptors (channel-invariant) ----
    if (tid < NPTS) {
        const float rx1 = rois[r * 5 + 1] * 0.25f;
        const float ry1 = rois[r * 5 + 2] * 0.25f;
        const float rx2 = rois[r * 5 + 3] * 0.25f;
        const float ry2 = rois[r * 5 + 4] * 0.25f;
        const float roi_w = fmaxf(rx2 - rx1, 1.0f);
        const float roi_h = fmaxf(ry2 - ry1, 1.0f);
        const float bin_w = roi_w * (1.0f / 7.0f);
        const float bin_h = roi_h * (1.0f / 7.0f);

        const int py = tid / 14;
        const int px = tid - py * 14;
        const float x = rx1 + ((float)px + 0.5f) * 0.5f * bin_w;
        const float y = ry1 + ((float)py + 0.5f) * 0.5f * bin_h;

        const bool valid = (y > -1.0f) && (y < (float)FH) &&
                           (x > -1.0f) && (x < (float)FW);
        const float yc = fminf(fmaxf(y, 0.0f), (float)(FH - 1));
        const float xc = fminf(fmaxf(x, 0.0f), (float)(FW - 1));
        const int y0 = (int)yc;               // yc >= 0 -> trunc == floor
        const int x0 = (int)xc;
        const int y1i = min(y0 + 1, FH - 1);
        const int x1i = min(x0 + 1, FW - 1);
        const float ly = yc - (float)y0;
        const float lx = xc - (float)x0;
        const float hy = 1.0f - ly;
        const float hx = 1.0f - lx;
        const float vm = valid ? 1.0f : 0.0f;

        s_off[tid] = make_int4(y0 * FW + x0, y0 * FW + x1i,
                               y1i * FW + x0, y1i * FW + x1i);
        s_w[tid] = make_float4(hy * hx * vm, hy * lx * vm,
                               ly * hx * vm, ly * lx * vm);
    }
    __syncthreads();

    // ---------------- Stage 2: gather + WMMA reduction --------------------------
    const int lane  = tid & 31;            // wave32
    const int wave  = tid >> 5;            // 0..7
    const int half  = lane >> 4;           // K-split half of the wave
    const int col   = lane & 15;           // A: channel row (M); B/D: bin column (N)
    const int cbase = (blockIdx.y * 8 + wave) * 16;    // 16-channel tile base
    const int batch = (int)rois[r * 5];

    // features[batch][cbase+col][.][.]
    const float* g = feat + (size_t)(batch * FC + cbase + col) * (FH * FW);

    v8f acc[4];
#pragma unroll
    for (int t = 0; t < 4; ++t) acc[t] = {};

    for (int k0 = 0; k0 < NPTS; k0 += 4) {
        const int p0 = k0 + half * 2;      // this half-wave's K index pair
        const int p1 = p0 + 1;

        // metadata broadcast within half-waves (LDS broadcast, no conflicts)
        const int4   o0 = s_off[p0];
        const float4 w0 = s_w[p0];
        const int4   o1 = s_off[p1];
        const float4 w1 = s_w[p1];

        // bilinear sample values for channel (cbase+col), points p0/p1
        const float a0 = w0.x * g[o0.x] + w0.y * g[o0.y] +
                         w0.z * g[o0.z] + w0.w * g[o0.w];
        const float a1 = w1.x * g[o1.x] + w1.y * g[o1.y] +
                         w1.z * g[o1.z] + w1.w * g[o1.w];
        v2f a;
        a.x = a0;                          // A VGPR0: K = half*2
        a.y = a1;                          // A VGPR1: K = half*2 + 1

        // static averaging matrix: bin(p) = (py>>1)*7 + (px>>1), weight 0.25
        const int b0 = ((p0 / 14) >> 1) * 7 + ((p0 % 14) >> 1);
        const int b1 = ((p1 / 14) >> 1) * 7 + ((p1 % 14) >> 1);

#pragma unroll
        for (int t = 0; t < 4; ++t) {
            const int nb = t * 16 + col;   // this lane's bin column in tile t
            v2f b;
            b.x = (b0 == nb) ? 0.25f : 0.0f;
            b.y = (b1 == nb) ? 0.25f : 0.0f;
            // D = A(16ch x 4pt) * B(4pt x 16bin) + C   (f32 throughout)
            acc[t] = __builtin_amdgcn_wmma_f32_16x16x4_f32(
                false, a, false, b, (short)0, acc[t], false, false);
        }
    }

    // ---------------- Store: D lane layout M = v + 8*half, N = col --------------
    const size_t obase = (size_t)r * FC * NBIN;
#pragma unroll
    for (int t = 0; t < 4; ++t) {
        const int bin = t * 16 + col;
        if (bin < NBIN) {
#pragma unroll
            for (int v = 0; v < 8; ++v) {
                const int ch = cbase + half * 8 + v;
                out[obase + (size_t)ch * NBIN + bin] = acc[t][v];
            }
        }
    }
}

extern "C" void kernel_launch(void* const* d_in, const int* in_sizes, int n_in,
                              void* d_out, int out_size, void* d_ws, size_t ws_size,
                              hipStream_t stream) {
    const float* feat = (const float*)d_in[0];   // [4,256,200,200] f32
    const float* rois = (const float*)d_in[1];   // [R,5] f32
    float* out = (float*)d_out;                  // [R,256,7,7] f32
    const int R = in_sizes[1] / 5;
    if (R <= 0) return;

    dim3 grid((unsigned)R, 2);                   // (roi, channel-half)
    dim3 block(256);                             // 8 wave32 waves, 1 tile each
    roialign_wmma_kernel<<<grid, block, 0, stream>>>(feat, rois, out);
}